// CNNadgnn_34479997452438
// MI455X (gfx1250) — compile-verified
//
#include <hip/hip_runtime.h>

#define NNODES 50000
#define NEDGES 800000
#define NB     32
#define HW     128
#define HW2    (HW*HW)
#define HID    128

static_assert(NNODES % 16 == 0, "tile loop assumes N is a multiple of 16");

typedef __attribute__((ext_vector_type(2))) float v2f;
typedef __attribute__((ext_vector_type(8))) float v8f;

// ---------------- 3x3 SAME conv, one pixel per thread, all OC per thread ----
template<int CIN, int OC, bool RELU>
__global__ void conv3x3_k(const float* __restrict__ in, const float* __restrict__ w,
                          const float* __restrict__ bias, float* __restrict__ out)
{
    __shared__ float wsh[OC*CIN*9];
    __shared__ float bsh[OC];
    for (int i = threadIdx.x; i < OC*CIN*9; i += blockDim.x) wsh[i] = w[i];
    if (threadIdx.x < OC) bsh[threadIdx.x] = bias[threadIdx.x];
    __syncthreads();

    int p = blockIdx.x * blockDim.x + threadIdx.x;
    if (p >= NB*HW2) return;
    int b = p >> 14;
    int y = (p >> 7) & (HW-1);
    int x = p & (HW-1);

    float acc[OC];
#pragma unroll
    for (int oc = 0; oc < OC; ++oc) acc[oc] = bsh[oc];

    for (int ic = 0; ic < CIN; ++ic) {
        int base = (b*CIN + ic) << 14;
#pragma unroll
        for (int dy = -1; dy <= 1; ++dy) {
            int yy = y + dy;
            if (yy < 0 || yy >= HW) continue;
            int rb = base + (yy << 7);
#pragma unroll
            for (int dx = -1; dx <= 1; ++dx) {
                int xx = x + dx;
                if (xx < 0 || xx >= HW) continue;
                float v = in[rb + xx];
                int wi = ic*9 + (dy+1)*3 + (dx+1);
#pragma unroll
                for (int oc = 0; oc < OC; ++oc)
                    acc[oc] += v * wsh[oc*CIN*9 + wi];
            }
        }
    }
#pragma unroll
    for (int oc = 0; oc < OC; ++oc) {
        float r = RELU ? fmaxf(acc[oc], 0.0f) : acc[oc];
        out[((b*OC + oc) << 14) + (y << 7) + x] = r;
    }
}

// ---------------- build graph input features gx = [x[:,0:2], x[:,5:8], pos, cnng]
__global__ void build_gx_k(const float* __restrict__ x, const float* __restrict__ pos,
                           const int* __restrict__ batch, const float* __restrict__ x1,
                           float* __restrict__ gx)
{
    int i = blockIdx.x*blockDim.x + threadIdx.x;
    if (i >= NNODES) return;
    float px = pos[2*i+0], py = pos[2*i+1];
    int c = (int)rintf(px * (1.0f/20.0f) * (float)(HW-1));
    int r = (int)rintf(py * (1.0f/20.0f) * (float)(HW-1));
    c = min(max(c, 0), HW-1);
    r = min(max(r, 0), HW-1);
    float cn = x1[(size_t)batch[i]*HW2 + r*HW + c];
    const float* xr = x + (size_t)i*8;
    float* g = gx + (size_t)i*8;
    g[0]=xr[0]; g[1]=xr[1]; g[2]=xr[5]; g[3]=xr[6]; g[4]=xr[7];
    g[5]=px;    g[6]=py;    g[7]=cn;
}

// ---------------- utility: zero fill ----------------
__global__ void zero_k(float* __restrict__ p, int n)
{
    int i = blockIdx.x*blockDim.x + threadIdx.x;
    if (i < n) p[i] = 0.0f;
}

// ---------------- in-degree (float) ----------------
__global__ void degree_k(const int* __restrict__ ei, float* __restrict__ deg)
{
    int e = blockIdx.x*blockDim.x + threadIdx.x;
    if (e >= NEDGES) return;
    atomicAdd(&deg[ei[NEDGES + e]], 1.0f);
}

// ---------------- edge scatter: agg[dst] += h[src], float4 granular -------
template<int DIN>
__global__ void scatter_k(const float* __restrict__ h, const int* __restrict__ ei,
                          float* __restrict__ agg)
{
    const int C4 = DIN/4;
    long long tid = (long long)blockIdx.x*blockDim.x + threadIdx.x;
    if (tid >= (long long)NEDGES*C4) return;
    int e = (int)(tid / C4);
    int c = ((int)(tid % C4)) << 2;
    int src = ei[e];
    int dst = ei[NEDGES + e];
    const float4 v = *(const float4*)(h + (size_t)src*DIN + c);
    float* a = agg + (size_t)dst*DIN + c;
    atomicAdd(a+0, v.x); atomicAdd(a+1, v.y);
    atomicAdd(a+2, v.z); atomicAdd(a+3, v.w);
}

// ---------------- fused normalize + dual GEMM via WMMA f32 16x16x4 --------
// out[n, :HID] = relu( (agg[n]/max(deg,1)) @ Wl^T + h[n] @ Wr^T + b )
// 256 threads = 8 wave32; wave w -> output cols [16w, 16w+16); block -> 16 nodes.
// Output staged through LDS so global stores are coalesced float4 (b128).
template<int DIN>
__global__ void sage_wmma_k(const float* __restrict__ agg, const float* __restrict__ h,
                            const float* __restrict__ deg,
                            const float* __restrict__ Wl, const float* __restrict__ Wr,
                            const float* __restrict__ bias, float* __restrict__ out)
{
    __shared__ __align__(16) float As[16][DIN];   // normalized neighbor mean
    __shared__ __align__(16) float Hs[16][DIN];   // self features
    __shared__ __align__(16) float Os[16][HID];   // staged output tile
    const int row0 = blockIdx.x * 16;             // always < NNODES, 16 | NNODES

    for (int idx = threadIdx.x; idx < 16*DIN; idx += 256) {
        int r = idx / DIN, c = idx % DIN;
        int g = row0 + r;
        float s = 1.0f / fmaxf(deg[g], 1.0f);
        As[r][c] = agg[(size_t)g*DIN + c] * s;
        Hs[r][c] = h[(size_t)g*DIN + c];
    }

    const int lane = threadIdx.x & 31;
    const int half = lane >> 4;      // 0: K pair {0,1}, 1: K pair {2,3}
    const int lr   = lane & 15;      // A: M row / B: N col within tile
    const int col0 = (threadIdx.x >> 5) << 4;

    // Warm L2/WGP$ for this wave's 16 weight rows (global_prefetch_b8).
    __builtin_prefetch(&Wl[(size_t)(col0 + lr)*DIN], 0, 0);
    __builtin_prefetch(&Wr[(size_t)(col0 + lr)*DIN], 0, 0);

    __syncthreads();

    v8f acc = {};
#pragma unroll
    for (int k = 0; k < DIN; k += 4) {
        int kk = k + 2*half;
        v2f a0 = *(const v2f*)&As[lr][kk];
        v2f b0 = *(const v2f*)&Wl[(size_t)(col0+lr)*DIN + kk];
        acc = __builtin_amdgcn_wmma_f32_16x16x4_f32(false, a0, false, b0,
                                                    (short)0, acc, false, false);
        v2f a1 = *(const v2f*)&Hs[lr][kk];
        v2f b1 = *(const v2f*)&Wr[(size_t)(col0+lr)*DIN + kk];
        acc = __builtin_amdgcn_wmma_f32_16x16x4_f32(false, a1, false, b1,
                                                    (short)0, acc, false, false);
    }

    // D layout: VGPR v holds element (M = v + 8*half, N = lr) of the tile.
    float bv = bias[col0 + lr];
#pragma unroll
    for (int v = 0; v < 8; ++v)
        Os[v + 8*half][col0 + lr] = fmaxf(acc[v] + bv, 0.0f);   // relu fused
    __syncthreads();

    // Coalesced b128 stores: 16*128 floats = 512 float4, 2 per thread.
    for (int idx = threadIdx.x; idx < 16*HID/4; idx += 256) {
        int r  = idx >> 5;          // / (HID/4)
        int c4 = (idx & 31) << 2;   // % (HID/4) * 4
        *(float4*)&out[(size_t)(row0 + r)*HID + c4] = *(const float4*)&Os[r][c4];
    }
}

// ---------------- final layer (dout = 1): plain dot products --------------
__global__ void sage_out_k(const float* __restrict__ agg, const float* __restrict__ h,
                           const float* __restrict__ deg,
                           const float* __restrict__ Wl, const float* __restrict__ Wr,
                           const float* __restrict__ bias, float* __restrict__ out)
{
    int i = blockIdx.x*blockDim.x + threadIdx.x;
    if (i >= NNODES) return;
    float s = 1.0f / fmaxf(deg[i], 1.0f);
    float acc = bias[0];
    for (int k = 0; k < HID; ++k)
        acc += agg[(size_t)i*HID + k]*s*Wl[k] + h[(size_t)i*HID + k]*Wr[k];
    out[i] = acc;
}

extern "C" void kernel_launch(void* const* d_in, const int* in_sizes, int n_in,
                              void* d_out, int out_size, void* d_ws, size_t ws_size,
                              hipStream_t stream)
{
    const float* xdata = (const float*)d_in[0];
    const float* x     = (const float*)d_in[1];
    const float* pos   = (const float*)d_in[2];
    const int*   ei    = (const int*)d_in[3];
    const int*   batch = (const int*)d_in[4];
    const float* w1 = (const float*)d_in[5];  const float* b1 = (const float*)d_in[6];
    const float* w2 = (const float*)d_in[7];  const float* b2 = (const float*)d_in[8];
    const float* w3 = (const float*)d_in[9];  const float* b3 = (const float*)d_in[10];
    const float* Wl1=(const float*)d_in[11];  const float* Wr1=(const float*)d_in[12]; const float* bs1=(const float*)d_in[13];
    const float* Wl2=(const float*)d_in[14];  const float* Wr2=(const float*)d_in[15]; const float* bs2=(const float*)d_in[16];
    const float* Wl3=(const float*)d_in[17];  const float* Wr3=(const float*)d_in[18]; const float* bs3=(const float*)d_in[19];
    const float* Wl4=(const float*)d_in[20];  const float* Wr4=(const float*)d_in[21]; const float* bs4=(const float*)d_in[22];

    float* out  = (float*)d_out;
    float* x1   = out;                    // (32,1,128,128) = 524288 floats
    float* gout = out + (size_t)NB*HW2;   // (N,1)

    // Workspace layout (floats). Conv region is reused by graph h buffers.
    float* ws  = (float*)d_ws;
    const size_t CONVSZ = (size_t)NB*16*HW2;      // 8,388,608
    float* h1  = ws;
    float* h2  = ws + CONVSZ;
    float* hb0 = ws;                               // reuse conv region
    float* hb1 = ws + (size_t)NNODES*HID;          // 6.4M..12.8M < 2*CONVSZ
    float* agg = ws + 2*CONVSZ;                    // N*HID
    float* gx  = agg + (size_t)NNODES*HID;         // N*8
    float* deg = gx  + (size_t)NNODES*8;           // N

    const int TPB = 256;
    const int convGrid = (NB*HW2)/TPB;             // 2048
    const int nodeGrid = (NNODES + TPB - 1)/TPB;
    const int tileGrid = NNODES/16;                // 3125 (exact)
    const int edgeGrid = (NEDGES + TPB - 1)/TPB;

    // CNN stage
    conv3x3_k<8,16,true>  <<<convGrid, TPB, 0, stream>>>(xdata, w1, b1, h1);
    conv3x3_k<16,16,true> <<<convGrid, TPB, 0, stream>>>(h1,    w2, b2, h2);
    conv3x3_k<16,1,false> <<<convGrid, TPB, 0, stream>>>(h2,    w3, b3, x1);

    // Node features
    build_gx_k<<<nodeGrid, TPB, 0, stream>>>(x, pos, batch, x1, gx);

    // Degrees (layer-invariant)
    zero_k<<<nodeGrid, TPB, 0, stream>>>(deg, NNODES);
    degree_k<<<edgeGrid, TPB, 0, stream>>>(ei, deg);

    // Layer 1: din = 8
    zero_k<<<(NNODES*8 + TPB-1)/TPB, TPB, 0, stream>>>(agg, NNODES*8);
    scatter_k<8><<<(unsigned)(((long long)NEDGES*2 + TPB-1)/TPB), TPB, 0, stream>>>(gx, ei, agg);
    sage_wmma_k<8><<<tileGrid, TPB, 0, stream>>>(agg, gx, deg, Wl1, Wr1, bs1, hb0);

    // Layer 2: din = 128
    zero_k<<<(NNODES*HID + TPB-1)/TPB, TPB, 0, stream>>>(agg, NNODES*HID);
    scatter_k<HID><<<(unsigned)(((long long)NEDGES*32 + TPB-1)/TPB), TPB, 0, stream>>>(hb0, ei, agg);
    sage_wmma_k<HID><<<tileGrid, TPB, 0, stream>>>(agg, hb0, deg, Wl2, Wr2, bs2, hb1);

    // Layer 3: din = 128
    zero_k<<<(NNODES*HID + TPB-1)/TPB, TPB, 0, stream>>>(agg, NNODES*HID);
    scatter_k<HID><<<(unsigned)(((long long)NEDGES*32 + TPB-1)/TPB), TPB, 0, stream>>>(hb1, ei, agg);
    sage_wmma_k<HID><<<tileGrid, TPB, 0, stream>>>(agg, hb1, deg, Wl3, Wr3, bs3, hb0);

    // Layer 4: din = 128, dout = 1
    zero_k<<<(NNODES*HID + TPB-1)/TPB, TPB, 0, stream>>>(agg, NNODES*HID);
    scatter_k<HID><<<(unsigned)(((long long)NEDGES*32 + TPB-1)/TPB), TPB, 0, stream>>>(hb0, ei, agg);
    sage_out_k<<<nodeGrid, TPB, 0, stream>>>(agg, hb0, deg, Wl4, Wr4, bs4, gout);
}